// SUB_Cross_Loss_44796508897296
// MI455X (gfx1250) — compile-verified
//
#include <hip/hip_runtime.h>
#include <hip/hip_bf16.h>

typedef __attribute__((ext_vector_type(2))) float v2f;
typedef __attribute__((ext_vector_type(8))) float v8f;

// ---------------------------------------------------------------------------
// Kernel 1: sim = A (M x K) * B^T (N x K)  -> C (M x N), fp32 via WMMA 16x16x4
// One wave per 16x16 output tile. A-fragment: lane l (l16 = l&15, half = l>>4)
// holds A[mi+l16][k + 2*half + {0,1}] (a contiguous float2). B-fragment is the
// symmetric layout on B rows (B = f2, row-major N x K, so B^T[k][n]=B[n][k]).
// ---------------------------------------------------------------------------
__global__ void gemm_wmma_f32(const float* __restrict__ A,
                              const float* __restrict__ B,
                              float* __restrict__ C,
                              int M, int N, int K) {
  const int lane = threadIdx.x & 31;
  const int wave = threadIdx.x >> 5;
  const int tilesN = N >> 4;
  const int tiles  = (M >> 4) * tilesN;
  const int tile = blockIdx.x * (blockDim.x >> 5) + wave;
  if (tile >= tiles) return;                 // wave-uniform: EXEC stays all-1s

  const int mi = (tile / tilesN) << 4;
  const int nj = (tile % tilesN) << 4;
  const int half = lane >> 4;                // 0: K={0,1}, 1: K={2,3}
  const int l16  = lane & 15;

  const float* aPtr = A + (size_t)(mi + l16) * K + 2 * half;
  const float* bPtr = B + (size_t)(nj + l16) * K + 2 * half;

  v8f c = {};
#pragma unroll 4
  for (int k = 0; k < K; k += 4) {
    v2f a = *(const v2f*)(aPtr + k);
    v2f b = *(const v2f*)(bPtr + k);
    // 8 args: (neg_a, A, neg_b, B, c_mod, C, reuse_a, reuse_b)
    c = __builtin_amdgcn_wmma_f32_16x16x4_f32(false, a, false, b,
                                              (short)0, c, false, false);
  }

  // C/D layout: VGPR v, lanes 0-15 -> (M = v,   N = lane),
  //                     lanes 16-31 -> (M = v+8, N = lane-16)
  float* cPtr = C + (size_t)(mi + 8 * half) * N + nj + l16;
#pragma unroll
  for (int v = 0; v < 8; ++v) cPtr[(size_t)v * N] = c[v];
}

// ---------------------------------------------------------------------------
// Kernel 2: per-row pairwise softplus loss. One block (256 thr) per row.
// Compacts positives/negatives of the row into LDS, then P x N loop.
// transposed=0: row i of sim (stride 1);  transposed=1: column i (stride ld).
// acc[0] += loss, acc[1] += npos*nneg (only if msel[row]).
// ---------------------------------------------------------------------------
#define MAX_COLS 512

__global__ void pair_loss_kernel(const float* __restrict__ sim,
                                 const int* __restrict__ mask,
                                 const unsigned char* __restrict__ msel,
                                 float* __restrict__ acc,
                                 int nCols, int ld, int transposed) {
  const int row = blockIdx.x;
  if (msel[row] == 0) return;

  __shared__ float spos[MAX_COLS];
  __shared__ float sneg[MAX_COLS];
  __shared__ int cnt[2];                  // cnt[0]=npos, cnt[1]=nneg
  __shared__ float red[256];

  const int tid = threadIdx.x;
  if (tid < 2) cnt[tid] = 0;
  __syncthreads();

  for (int n = tid; n < nCols; n += blockDim.x) {
    const size_t idx = transposed ? ((size_t)n * ld + row)
                                  : ((size_t)row * ld + n);
    const float s = sim[idx];
    if (mask[idx] != 0) {
      spos[atomicAdd(&cnt[0], 1)] = s;
    } else {
      sneg[atomicAdd(&cnt[1], 1)] = s;
    }
  }
  __syncthreads();

  const int npos = cnt[0];
  const int nneg = cnt[1];

  float lacc = 0.0f;
  for (int p = 0; p < npos; ++p) {
    const float sp = spos[p];
    for (int n = tid; n < nneg; n += blockDim.x) {
      const float d = sneg[n] - sp;
      // numerically stable softplus(d) = max(d,0) + log1p(exp(-|d|))
      lacc += fmaxf(d, 0.0f) + log1pf(expf(-fabsf(d)));
    }
  }

  red[tid] = lacc;
  __syncthreads();
  for (int s = 128; s > 0; s >>= 1) {
    if (tid < s) red[tid] += red[tid + s];
    __syncthreads();
  }
  if (tid == 0) {
    atomicAdd(&acc[0], red[0]);
    atomicAdd(&acc[1], (float)npos * (float)nneg);
  }
}

// ---------------------------------------------------------------------------
// Init / finalize
// ---------------------------------------------------------------------------
__global__ void init_acc_kernel(float* __restrict__ acc) {
  if (threadIdx.x < 2) acc[threadIdx.x] = 0.0f;
}

__global__ void finalize_kernel(const float* __restrict__ acc,
                                float* __restrict__ out) {
  out[0] = (acc[1] > 0.0f) ? (acc[0] / acc[1]) : acc[0];
}

// ---------------------------------------------------------------------------
extern "C" void kernel_launch(void* const* d_in, const int* in_sizes, int n_in,
                              void* d_out, int out_size, void* d_ws, size_t ws_size,
                              hipStream_t stream) {
  const float* f1 = (const float*)d_in[0];            // [N1, D]
  const float* f2 = (const float*)d_in[1];            // [N2, D]
  const int* mask = (const int*)d_in[2];              // [N1, N2]
  const unsigned char* ms1 = (const unsigned char*)d_in[3];  // [N1] bool
  const unsigned char* ms2 = (const unsigned char*)d_in[4];  // [N2] bool

  const int N1 = in_sizes[3];
  const int N2 = in_sizes[4];
  const int D  = in_sizes[0] / N1;

  float* acc = (float*)d_ws;                          // 2 floats
  float* sim = (float*)((char*)d_ws + 256);           // [N1, N2] fp32

  init_acc_kernel<<<1, 32, 0, stream>>>(acc);

  const int tiles = (N1 / 16) * (N2 / 16);
  const int wavesPerBlock = 8;                        // 256 threads, wave32
  gemm_wmma_f32<<<(tiles + wavesPerBlock - 1) / wavesPerBlock, 256, 0, stream>>>(
      f1, f2, sim, N1, N2, D);

  // Direction 1: rows of sim, mask rows, ms1
  pair_loss_kernel<<<N1, 256, 0, stream>>>(sim, mask, ms1, acc, N2, N2, 0);
  // Direction 2: columns of sim, mask columns, ms2
  pair_loss_kernel<<<N2, 256, 0, stream>>>(sim, mask, ms2, acc, N1, N2, 1);

  finalize_kernel<<<1, 1, 0, stream>>>(acc, (float*)d_out);
}